// SpatialTransformer_8220567405308
// MI455X (gfx1250) — compile-verified
//
#include <hip/hip_runtime.h>
#include <stdint.h>

#define H_IN   192
#define W_IN   192
#define H_OUT  96
#define W_OUT  96
#define ROWS_PER_BLK 16
#define THREADS 256
#define MAX_R  32
#define TILE_ELEMS (MAX_R * 192)   // 24 KB worst case

typedef int v4i __attribute__((ext_vector_type(4)));
typedef __attribute__((address_space(1))) v4i* global_v4i_ptr;
typedef __attribute__((address_space(3))) v4i* lds_v4i_ptr;

// ---- CDNA5 async global->LDS copy (ASYNCcnt path), with asm fallback ----
#if defined(__HIP_DEVICE_COMPILE__) && __has_builtin(__builtin_amdgcn_global_load_async_to_lds_b128)
#define USE_ASYNC_BUILTIN 1
#else
#define USE_ASYNC_BUILTIN 0
#endif

#if defined(__HIP_DEVICE_COMPILE__) && __has_builtin(__builtin_amdgcn_s_wait_asynccnt)
#define USE_WAIT_BUILTIN 1
#else
#define USE_WAIT_BUILTIN 0
#endif

__device__ __forceinline__ void async_copy_b128(const float* gp, float* lp) {
#if USE_ASYNC_BUILTIN
  __builtin_amdgcn_global_load_async_to_lds_b128(
      (global_v4i_ptr)gp,   // C-style cast: drops const, retargets to AS1, retypes to v4i
      (lds_v4i_ptr)lp,
      0, 0);
#else
  unsigned l = (unsigned)(size_t)(__attribute__((address_space(3))) float*)lp;
  asm volatile("global_load_async_to_lds_b128 %0, %1, off"
               :: "v"(l), "v"(gp) : "memory");
#endif
}

__device__ __forceinline__ void wait_async_zero() {
#if USE_WAIT_BUILTIN
  __builtin_amdgcn_s_wait_asynccnt(0);
#else
  asm volatile("s_wait_asynccnt 0x0" ::: "memory");
#endif
}

__global__ __launch_bounds__(THREADS)
void st_bilinear_kernel(const float* __restrict__ x,
                        const float* __restrict__ zw,
                        float* __restrict__ out) {
  __shared__ float tile[TILE_ELEMS];

  const int b  = blockIdx.y;
  const int h0 = blockIdx.x * ROWS_PER_BLK;

  // Block-uniform transform params -> scalar loads
  const float s  = zw[b * 3 + 0];
  const float tx = zw[b * 3 + 1];
  const float ty = zw[b * 3 + 2];

  // ---- Input row span needed for output rows [h0, h0+15] ----
  const float ysA = 2.0f * ((float)h0 + 0.5f) * (1.0f / H_OUT) - 1.0f;
  const float ysB = 2.0f * ((float)h0 + (float)ROWS_PER_BLK - 0.5f) * (1.0f / H_OUT) - 1.0f;
  const float iyA = (s * ysA + ty + 1.0f) * (0.5f * H_IN) - 0.5f;
  const float iyB = (s * ysB + ty + 1.0f) * (0.5f * H_IN) - 0.5f;
  int r0 = (int)floorf(iyA);
  int r1 = (int)floorf(iyB) + 1;
  r0 = min(max(r0, 0), H_IN - 1);
  r1 = min(max(r1, r0), H_IN - 1);
  int R = r1 - r0 + 1;
  if (R > MAX_R) R = MAX_R;   // cannot trigger for s<=1; memory-safety clamp

  // ---- Input column span (uniform across all output rows) ----
  const float xsA = 2.0f * 0.5f * (1.0f / W_OUT) - 1.0f;
  const float xsB = 2.0f * ((float)W_OUT - 0.5f) * (1.0f / W_OUT) - 1.0f;
  const float ixA = (s * xsA + tx + 1.0f) * (0.5f * W_IN) - 0.5f;
  const float ixB = (s * xsB + tx + 1.0f) * (0.5f * W_IN) - 0.5f;
  int c0 = (int)floorf(ixA);
  int c1 = (int)floorf(ixB) + 1;
  c0 = min(max(c0, 0), W_IN - 1);
  c0 &= ~31;                              // 128B-aligned start, <= 160
  c1 = min(max(c1, c0), W_IN - 1);
  const int W32 = (c1 - c0 + 32) & ~31;   // padded width; c0+W32-1 <= 191 always

  // ---- Stage rectangle [r0, r0+R) x [c0, c0+W32) via async B128 bursts ----
  const float* gbase = x + (size_t)b * (H_IN * W_IN);
  const int cpr = W32 >> 2;               // 16-byte chunks per row
  const int nChunks = R * cpr;
  for (int i = threadIdx.x; i < nChunks; i += THREADS) {
    const int rr = i / cpr;
    const int cc = (i - rr * cpr) << 2;
    async_copy_b128(gbase + (size_t)(r0 + rr) * W_IN + (c0 + cc),
                    &tile[rr * W32 + cc]);
  }
  wait_async_zero();
  __syncthreads();

  // ---- Sample 16x96 output pixels from LDS ----
  for (int idx = threadIdx.x; idx < ROWS_PER_BLK * W_OUT; idx += THREADS) {
    const int h = idx / W_OUT;
    const int w = idx - h * W_OUT;

    const float ys = 2.0f * ((float)(h0 + h) + 0.5f) * (1.0f / H_OUT) - 1.0f;
    const float xs = 2.0f * ((float)w + 0.5f) * (1.0f / W_OUT) - 1.0f;
    const float iy = (s * ys + ty + 1.0f) * (0.5f * H_IN) - 0.5f;
    const float ix = (s * xs + tx + 1.0f) * (0.5f * W_IN) - 0.5f;

    const float iy0f = floorf(iy);
    const float ix0f = floorf(ix);
    const float wy1 = iy - iy0f, wy0 = 1.0f - wy1;
    const float wx1 = ix - ix0f, wx0 = 1.0f - wx1;
    const int iy0 = (int)iy0f, ix0 = (int)ix0f;
    const int iy1 = iy0 + 1,   ix1 = ix0 + 1;

    const bool vy0 = (iy0 >= 0) && (iy0 < H_IN);
    const bool vy1 = (iy1 >= 0) && (iy1 < H_IN);
    const bool vx0 = (ix0 >= 0) && (ix0 < W_IN);
    const bool vx1 = (ix1 >= 0) && (ix1 < W_IN);

    // Clamped LDS offsets (always in-bounds); masked to zero when invalid.
    const int ly0 = min(max(iy0 - r0, 0), R - 1) * W32;
    const int ly1 = min(max(iy1 - r0, 0), R - 1) * W32;
    const int lx0 = min(max(ix0 - c0, 0), W32 - 1);
    const int lx1 = min(max(ix1 - c0, 0), W32 - 1);

    const float v00 = tile[ly0 + lx0] * ((vy0 && vx0) ? 1.0f : 0.0f);
    const float v01 = tile[ly0 + lx1] * ((vy0 && vx1) ? 1.0f : 0.0f);
    const float v10 = tile[ly1 + lx0] * ((vy1 && vx0) ? 1.0f : 0.0f);
    const float v11 = tile[ly1 + lx1] * ((vy1 && vx1) ? 1.0f : 0.0f);

    const float res = wy0 * (wx0 * v00 + wx1 * v01) + wy1 * (wx0 * v10 + wx1 * v11);

    // Output is write-once, never re-read: bypass near caches.
    __builtin_nontemporal_store(res, &out[((size_t)b * H_OUT + (h0 + h)) * W_OUT + w]);
  }
}

extern "C" void kernel_launch(void* const* d_in, const int* in_sizes, int n_in,
                              void* d_out, int out_size, void* d_ws, size_t ws_size,
                              hipStream_t stream) {
  const float* x  = (const float*)d_in[0];   // (B,1,192,192) f32
  const float* zw = (const float*)d_in[1];   // (B,3) f32
  float* out = (float*)d_out;                // (B,1,96,96) f32

  const int B = in_sizes[1] / 3;
  dim3 grid(H_OUT / ROWS_PER_BLK, B);        // (6, 1024)
  dim3 block(THREADS);
  st_bilinear_kernel<<<grid, block, 0, stream>>>(x, zw, out);
}